// SingleReadoutModel_15358803050683
// MI455X (gfx1250) — compile-verified
//
#include <hip/hip_runtime.h>
#include <math.h>

// Problem constants (from reference)
#define NN 10000      // nodes
#define NE 160000     // edges
#define NK 4          // species
#define NC 64         // channels
#define NB 8          // radial bessel count
#define NH 64         // radial hidden
#define NL 2          // layers
#define NG 50         // graphs
#define NS 16         // spherical-harmonic components (l<=3)
#define FEAT (NC*NS)  // 1024 per node
#define RMAXF 5.0f
#define INV_AVG 0.0625f   // 1/16 avg-neighbor normalization

typedef float v2f __attribute__((ext_vector_type(2)));
typedef float v8f __attribute__((ext_vector_type(8)));

__device__ __forceinline__ v8f wmma4(v2f a, v2f b, v8f c) {
  // D = A(16x4,f32) * B(4x16,f32) + C(16x16,f32)
  return __builtin_amdgcn_wmma_f32_16x16x4_f32(false, a, false, b, (short)0, c, false, false);
}

__device__ __forceinline__ float fast_sigmoid(float x) {
  // single v_rcp_f32 instead of IEEE division chain (~1 ulp, fine for gating)
  return __builtin_amdgcn_rcpf(1.f + __expf(-x));
}

#define V8_ZERO {0.f,0.f,0.f,0.f,0.f,0.f,0.f,0.f}

// sph constants
#define S3   1.7320508075688772f
#define S5   2.2360679774997896f
#define S15  3.8729833462074170f
#define S7   2.6457513110645907f
#define S105 10.246950765959598f
#define S358 2.0916500663351889f   // sqrt(35/8)
#define S218 1.6201851746019651f   // sqrt(21/8)

// ---------------------------------------------------------------- utilities
__global__ void k_zero(float* __restrict__ p, long n) {
  long t = (long)blockIdx.x * blockDim.x + threadIdx.x;
  if (t < n) p[t] = 0.f;
}

// ------------------------------------------------------- edge geometry (fwd)
__global__ void k_edge_geom(const float* __restrict__ pos,
                            const float* __restrict__ shifts,
                            const int* __restrict__ ei,
                            float* __restrict__ r_, float* __restrict__ u_,
                            float* __restrict__ Y, float* __restrict__ Rb,
                            float* __restrict__ dRb) {
  int e = blockIdx.x * blockDim.x + threadIdx.x;
  if (e >= NE) return;
  int s = ei[e], t = ei[NE + e];
  float vx = pos[t*3+0] - pos[s*3+0] + shifts[e*3+0];
  float vy = pos[t*3+1] - pos[s*3+1] + shifts[e*3+1];
  float vz = pos[t*3+2] - pos[s*3+2] + shifts[e*3+2];
  float r = sqrtf(vx*vx + vy*vy + vz*vz + 1e-12f);
  float inv = 1.f / r;
  float x = vx*inv, y = vy*inv, z = vz*inv;
  r_[e] = r; u_[e*3+0] = x; u_[e*3+1] = y; u_[e*3+2] = z;

  float* Ye = Y + (long)e*NS;
  Ye[0] = 1.f;
  Ye[1] = S3*x;  Ye[2] = S3*y;  Ye[3] = S3*z;
  Ye[4] = S15*x*y; Ye[5] = S15*y*z; Ye[6] = 0.5f*S5*(3.f*z*z-1.f);
  Ye[7] = S15*x*z; Ye[8] = 0.5f*S15*(x*x-y*y);
  Ye[9]  = S358*y*(3.f*x*x-y*y);
  Ye[10] = S105*x*y*z;
  Ye[11] = S218*y*(5.f*z*z-1.f);
  Ye[12] = 0.5f*S7*(5.f*z*z*z-3.f*z);
  Ye[13] = S218*x*(5.f*z*z-1.f);
  Ye[14] = 0.5f*S105*z*(x*x-y*y);
  Ye[15] = S358*x*(x*x-3.f*y*y);

  // radial: bessel * polynomial cutoff (p=6): env = 1 - 28u^6 + 48u^7 - 21u^8
  float uu = r / RMAXF;
  float env = 0.f, denv = 0.f;
  if (uu < 1.f) {
    float u2 = uu*uu, u4 = u2*u2, u5 = u4*uu, u6 = u5*uu, u7 = u6*uu, u8 = u7*uu;
    env  = 1.f - 28.f*u6 + 48.f*u7 - 21.f*u8;
    denv = (-168.f*u5 + 336.f*u6 - 168.f*u7) / RMAXF;
  }
  const float c0 = 0.63245553203367587f; // sqrt(2/5)
  #pragma unroll
  for (int n = 1; n <= NB; n++) {
    float w  = (float)n * 3.14159265358979323846f / RMAXF;
    float sn = __sinf(w*r), cs = __cosf(w*r);
    float bess  = c0 * sn * inv;
    float dbess = c0 * (w*cs*inv - sn*inv*inv);
    Rb [(long)e*NB + n-1] = bess*env;
    dRb[(long)e*NB + n-1] = dbess*env + bess*denv;
  }
}

// ------------------------------------------------------------ node embedding
__global__ void k_embed(const float* __restrict__ attrs, const float* __restrict__ Wemb,
                        float* __restrict__ f0, int* __restrict__ spec) {
  long t = (long)blockIdx.x * blockDim.x + threadIdx.x;
  if (t >= (long)NN*NC) return;
  int n = (int)(t >> 6), c = (int)(t & 63);
  float a0 = attrs[n*NK+0], a1 = attrs[n*NK+1], a2 = attrs[n*NK+2], a3 = attrs[n*NK+3];
  float emb = a0*Wemb[c] + a1*Wemb[NC+c] + a2*Wemb[2*NC+c] + a3*Wemb[3*NC+c];
  float* f = f0 + (long)n*FEAT + c*NS;
  f[0] = emb;
  #pragma unroll
  for (int s = 1; s < NS; s++) f[s] = 0.f;
  if (c == 0) {
    int k = 0; float m = a0;
    if (a1 > m) { m = a1; k = 1; }
    if (a2 > m) { m = a2; k = 2; }
    if (a3 > m) { m = a3; k = 3; }
    spec[n] = k;
  }
}

// ------------------------------------- radial MLP forward: Re = silu(Rb@W1)@W2
// 8 waves/block, each wave owns a 16-edge tile. WMMA f32 16x16x4.
__global__ void __launch_bounds__(256) k_radial_fwd(const float* __restrict__ Rb,
                        const float* __restrict__ rW1l, const float* __restrict__ rW2l,
                        float* __restrict__ Re) {
  __shared__ float h1[8*16*64];
  int wid = threadIdx.x >> 5, lane = threadIdx.x & 31;
  int lo = lane & 15, hi = lane >> 4;
  int e0 = (blockIdx.x*8 + wid)*16;
  float* hw = h1 + wid*(16*64);
  int ea = min(e0 + lo, NE-1);
  // pre = Rb(16x8) @ rW1(8x64); h1 = silu(pre)
  #pragma unroll
  for (int nt = 0; nt < 4; nt++) {
    v8f acc = V8_ZERO;
    #pragma unroll
    for (int k = 0; k < NB; k += 4) {
      v2f a, b;
      a.x = Rb[(long)ea*NB + k + 2*hi];
      a.y = Rb[(long)ea*NB + k + 2*hi + 1];
      b.x = rW1l[(k + 2*hi)*NH + nt*16 + lo];
      b.y = rW1l[(k + 2*hi + 1)*NH + nt*16 + lo];
      acc = wmma4(a, b, acc);
    }
    #pragma unroll
    for (int i = 0; i < 8; i++) {
      float p = acc[i];
      hw[(i + 8*hi)*64 + nt*16 + lo] = p * fast_sigmoid(p);
    }
  }
  __syncthreads();
  // Re = h1(16x64) @ rW2(64x64)
  #pragma unroll
  for (int nt = 0; nt < 4; nt++) {
    v8f acc = V8_ZERO;
    for (int k = 0; k < NH; k += 4) {
      v2f a, b;
      a.x = hw[lo*64 + k + 2*hi];
      a.y = hw[lo*64 + k + 2*hi + 1];
      b.x = rW2l[(k + 2*hi)*NC + nt*16 + lo];
      b.y = rW2l[(k + 2*hi + 1)*NC + nt*16 + lo];
      acc = wmma4(a, b, acc);
    }
    #pragma unroll
    for (int i = 0; i < 8; i++) {
      int e = e0 + i + 8*hi;
      if (e < NE) Re[(long)e*NC + nt*16 + lo] = acc[i];
    }
  }
}

// --------------------------- message + scatter: agg[r,c,s] += gate*Y/avg_nei
__global__ void k_msg_fwd(const float* __restrict__ fin, const float* __restrict__ Re,
                          const float* __restrict__ Y, const int* __restrict__ ei,
                          float* __restrict__ agg) {
  long t = (long)blockIdx.x * blockDim.x + threadIdx.x;
  if (t >= (long)NE*NC) return;
  int e = (int)(t >> 6), c = (int)(t & 63);
  int s = ei[e], rv = ei[NE + e];
  float gate = Re[(long)e*NC + c] * fin[(long)s*FEAT + c*NS] * INV_AVG;
  const float* Ye = Y + (long)e*NS;
  float* ab = agg + (long)rv*FEAT + c*NS;
  #pragma unroll
  for (int q = 0; q < NS; q++) atomicAdd(ab + q, gate * Ye[q]);
}

// ------- node update: fout[d,s] = sum_c agg[c,s]*Wmix[c,d] + fin[c,s]*Wsk[c,d]
// computed as out^T = X^T * W; one block per node, 4 waves = 4 d-tiles.
__global__ void __launch_bounds__(128) k_node_update(const float* __restrict__ fin,
        const float* __restrict__ agg, const float* __restrict__ Wmix,
        const float* __restrict__ Wskip, const int* __restrict__ spec,
        float* __restrict__ fout) {
  int n = blockIdx.x;
  int wid = threadIdx.x >> 5, lane = threadIdx.x & 31;
  int lo = lane & 15, hi = lane >> 4;
  const float* Wsk = Wskip + (long)spec[n]*(NC*NC);
  __builtin_prefetch(Wsk, 0, 3);
  const float* A1 = agg + (long)n*FEAT;
  const float* A2 = fin + (long)n*FEAT;
  v8f acc = V8_ZERO;
  for (int k = 0; k < NC; k += 4) {
    v2f a, b;
    a.x = A1[(k + 2*hi)*NS + lo];   a.y = A1[(k + 2*hi + 1)*NS + lo];
    b.x = Wmix[(k + 2*hi)*NC + wid*16 + lo];
    b.y = Wmix[(k + 2*hi + 1)*NC + wid*16 + lo];
    acc = wmma4(a, b, acc);
  }
  for (int k = 0; k < NC; k += 4) {
    v2f a, b;
    a.x = A2[(k + 2*hi)*NS + lo];   a.y = A2[(k + 2*hi + 1)*NS + lo];
    b.x = Wsk[(k + 2*hi)*NC + wid*16 + lo];
    b.y = Wsk[(k + 2*hi + 1)*NC + wid*16 + lo];
    acc = wmma4(a, b, acc);
  }
  float* fo = fout + (long)n*FEAT;
  #pragma unroll
  for (int i = 0; i < 8; i++) fo[(wid*16 + lo)*NS + (i + 8*hi)] = acc[i];
}

// ------------------------------------------------------------------- readout
__global__ void k_readout(const float* __restrict__ f2, const float* __restrict__ attrs,
                          const float* __restrict__ ae, const float* __restrict__ wro,
                          const int* __restrict__ batch, float* __restrict__ out) {
  int n = blockIdx.x * blockDim.x + threadIdx.x;
  if (n >= NN) return;
  float e = 0.f;
  const float* f = f2 + (long)n*FEAT;
  for (int c = 0; c < NC; c++) e += f[c*NS] * wro[c];
  #pragma unroll
  for (int k = 0; k < NK; k++) e += attrs[n*NK + k] * ae[k];
  atomicAdd(&out[batch[n]], e);
}

// ------------------------------------------------- backward seed: dE/dfeats2
__global__ void k_init_grad(const float* __restrict__ wro, float* __restrict__ gF) {
  long t = (long)blockIdx.x * blockDim.x + threadIdx.x;
  if (t >= (long)NN*NC) return;
  int n = (int)(t >> 6), c = (int)(t & 63);
  float* g = gF + (long)n*FEAT + c*NS;
  g[0] = wro[c];
  #pragma unroll
  for (int s = 1; s < NS; s++) g[s] = 0.f;
}

// ------- node update backward: gAgg = Wmix @ gOut, gIn = Wsk @ gOut (per node)
__global__ void __launch_bounds__(128) k_node_update_bwd(const float* __restrict__ gOut,
        const float* __restrict__ Wmix, const float* __restrict__ Wskip,
        const int* __restrict__ spec,
        float* __restrict__ gAgg, float* __restrict__ gIn) {
  int n = blockIdx.x;
  int wid = threadIdx.x >> 5, lane = threadIdx.x & 31;
  int lo = lane & 15, hi = lane >> 4;
  const float* Wsk = Wskip + (long)spec[n]*(NC*NC);
  const float* G = gOut + (long)n*FEAT;
  v8f acc1 = V8_ZERO, acc2 = V8_ZERO;
  for (int k = 0; k < NC; k += 4) {
    v2f a1, a2, b;
    b.x = G[(k + 2*hi)*NS + lo];    b.y = G[(k + 2*hi + 1)*NS + lo];
    a1.x = Wmix[(wid*16 + lo)*NC + k + 2*hi];
    a1.y = Wmix[(wid*16 + lo)*NC + k + 2*hi + 1];
    a2.x = Wsk[(wid*16 + lo)*NC + k + 2*hi];
    a2.y = Wsk[(wid*16 + lo)*NC + k + 2*hi + 1];
    acc1 = wmma4(a1, b, acc1);
    acc2 = wmma4(a2, b, acc2);
  }
  float* ga = gAgg + (long)n*FEAT;
  float* gi = gIn  + (long)n*FEAT;
  #pragma unroll
  for (int i = 0; i < 8; i++) {
    int c = wid*16 + i + 8*hi;
    ga[c*NS + lo] = acc1[i];
    gi[c*NS + lo] = acc2[i];
  }
}

// ----------- message backward: g_gate, g_Re, g_Y, and sender-scalar gradient
__global__ void __launch_bounds__(256) k_msg_bwd(const float* __restrict__ fin,
        const float* __restrict__ Re, const float* __restrict__ Y,
        const float* __restrict__ gAgg, const int* __restrict__ ei,
        float* __restrict__ gIn, float* __restrict__ gRe, float* __restrict__ gY) {
  __shared__ float sY[4][NS];
  int c = threadIdx.x & 63, el = threadIdx.x >> 6;
  int e = blockIdx.x*4 + el;
  if (threadIdx.x < 64) sY[threadIdx.x >> 4][threadIdx.x & 15] = 0.f;
  __syncthreads();
  if (e < NE) {
    int s = ei[e], rv = ei[NE + e];
    float f0 = fin[(long)s*FEAT + c*NS];
    float rc = Re[(long)e*NC + c];
    float gate = rc * f0;
    const float* ga = gAgg + (long)rv*FEAT + c*NS;
    const float* Ye = Y + (long)e*NS;
    float gg = 0.f;
    #pragma unroll
    for (int q = 0; q < NS; q++) {
      float gm = ga[q] * INV_AVG;
      gg += gm * Ye[q];
      atomicAdd(&sY[el][q], gm * gate);     // LDS ds_add_f32 reduction over c
    }
    gRe[(long)e*NC + c] = gg * f0;
    atomicAdd(&gIn[(long)s*FEAT + c*NS], gg * rc);
  }
  __syncthreads();
  if (e < NE && c < NS) gY[(long)e*NS + c] += sY[el][c];  // block owns edge e
}

// ----- radial backward: g_h1 = gRe @ W2^T (WMMA); g_pre = g_h1*silu'(pre);
//       gRb += g_pre @ W1^T.  pre recomputed (K=8, cheap) instead of stored.
__global__ void __launch_bounds__(256) k_radial_bwd(const float* __restrict__ gRe,
        const float* __restrict__ Rb, const float* __restrict__ rW1l,
        const float* __restrict__ rW2l, float* __restrict__ gRb) {
  __shared__ float gp[8*16*64];
  int wid = threadIdx.x >> 5, lane = threadIdx.x & 31;
  int lo = lane & 15, hi = lane >> 4;
  int e0 = (blockIdx.x*8 + wid)*16;
  float* gw = gp + wid*1024;
  int ea = min(e0 + lo, NE-1);
  #pragma unroll
  for (int nt = 0; nt < 4; nt++) {
    v8f acc = V8_ZERO;
    for (int k = 0; k < NC; k += 4) {
      v2f a, b;
      a.x = gRe[(long)ea*NC + k + 2*hi];
      a.y = gRe[(long)ea*NC + k + 2*hi + 1];
      b.x = rW2l[(nt*16 + lo)*NC + k + 2*hi];      // W2^T as B operand
      b.y = rW2l[(nt*16 + lo)*NC + k + 2*hi + 1];
      acc = wmma4(a, b, acc);
    }
    #pragma unroll
    for (int i = 0; i < 8; i++) {
      int m = i + 8*hi, h = nt*16 + lo;
      int e = min(e0 + m, NE-1);
      float pre = 0.f;
      #pragma unroll
      for (int b2 = 0; b2 < NB; b2++) pre += Rb[(long)e*NB + b2] * rW1l[b2*NH + h];
      float sg = fast_sigmoid(pre);
      float sact = sg * (1.f + pre * (1.f - sg));  // silu'
      gw[m*64 + h] = acc[i] * sact;
    }
  }
  __syncthreads();
  #pragma unroll
  for (int j = 0; j < 4; j++) {
    int slot = lane*4 + j;
    int m = slot >> 3, b = slot & 7;
    float sum = 0.f;
    for (int h = 0; h < NH; h++) sum += gw[m*64 + h] * rW1l[b*NH + h];
    int e = e0 + m;
    if (e < NE) gRb[(long)e*NB + b] += sum;
  }
}

// ---------------- forces: chain rule through Y(u), u=vec/r, and radial basis
__global__ void k_force(const float* __restrict__ u_, const float* __restrict__ r_,
                        const float* __restrict__ gY, const float* __restrict__ gRb,
                        const float* __restrict__ dRb, const int* __restrict__ ei,
                        float* __restrict__ out) {
  int e = blockIdx.x * blockDim.x + threadIdx.x;
  if (e >= NE) return;
  float x = u_[e*3], y = u_[e*3+1], z = u_[e*3+2], r = r_[e];
  const float* g = gY + (long)e*NS;
  float gx = 0.f, gy = 0.f, gz = 0.f;
  gx += g[1]*S3;  gy += g[2]*S3;  gz += g[3]*S3;
  gx += g[4]*S15*y;  gy += g[4]*S15*x;
  gy += g[5]*S15*z;  gz += g[5]*S15*y;
  gz += g[6]*3.f*S5*z;
  gx += g[7]*S15*z;  gz += g[7]*S15*x;
  gx += g[8]*S15*x;  gy -= g[8]*S15*y;
  gx += g[9]*S358*6.f*x*y;  gy += g[9]*S358*3.f*(x*x - y*y);
  gx += g[10]*S105*y*z;  gy += g[10]*S105*x*z;  gz += g[10]*S105*x*y;
  float z5 = 5.f*z*z - 1.f;
  gy += g[11]*S218*z5;  gz += g[11]*S218*10.f*y*z;
  gz += g[12]*0.5f*S7*(15.f*z*z - 3.f);
  gx += g[13]*S218*z5;  gz += g[13]*S218*10.f*x*z;
  gx += g[14]*S105*x*z;  gy -= g[14]*S105*y*z;  gz += g[14]*0.5f*S105*(x*x - y*y);
  gx += g[15]*S358*3.f*(x*x - y*y);  gy -= g[15]*S358*6.f*x*y;

  float gr = 0.f;
  #pragma unroll
  for (int b = 0; b < NB; b++) gr += gRb[(long)e*NB + b] * dRb[(long)e*NB + b];

  float dot = gx*x + gy*y + gz*z;
  float inv = 1.f / r;
  float vx = (gx - dot*x)*inv + gr*x;   // dE/dvec
  float vy = (gy - dot*y)*inv + gr*y;
  float vz = (gz - dot*z)*inv + gr*z;
  int s = ei[e], t = ei[NE + e];
  // vec = pos[t] - pos[s]; F = -dE/dpos
  atomicAdd(&out[NG + t*3+0], -vx); atomicAdd(&out[NG + t*3+1], -vy); atomicAdd(&out[NG + t*3+2], -vz);
  atomicAdd(&out[NG + s*3+0],  vx); atomicAdd(&out[NG + s*3+1],  vy); atomicAdd(&out[NG + s*3+2],  vz);
}

// ============================================================================
extern "C" void kernel_launch(void* const* d_in, const int* in_sizes, int n_in,
                              void* d_out, int out_size, void* d_ws, size_t ws_size,
                              hipStream_t stream) {
  (void)in_sizes; (void)n_in; (void)out_size; (void)ws_size;
  const float* pos    = (const float*)d_in[0];
  const float* attrs  = (const float*)d_in[1];
  const float* shifts = (const float*)d_in[2];
  const int*   ei     = (const int*)d_in[3];
  const int*   batch  = (const int*)d_in[4];
  const float* ae     = (const float*)d_in[6];
  const float* Wemb   = (const float*)d_in[7];
  const float* rW1    = (const float*)d_in[8];
  const float* rW2    = (const float*)d_in[9];
  const float* Wmix   = (const float*)d_in[10];
  const float* Wskip  = (const float*)d_in[11];
  const float* wro    = (const float*)d_in[12];
  float* out = (float*)d_out;

  float* w = (float*)d_ws;
  size_t o = 0;
  float* r_  = w + o; o += (size_t)NE;
  float* u_  = w + o; o += (size_t)3*NE;
  float* Y   = w + o; o += (size_t)16*NE;
  float* Rb  = w + o; o += (size_t)8*NE;
  float* dRb = w + o; o += (size_t)8*NE;
  float* Re0 = w + o; o += (size_t)64*NE;
  float* Re1 = w + o; o += (size_t)64*NE;
  float* gRe = w + o; o += (size_t)64*NE;
  float* gY  = w + o; o += (size_t)16*NE;   // gY..gRb contiguous (zeroed together)
  float* gRb = w + o; o += (size_t)8*NE;
  float* f0  = w + o; o += (size_t)NN*FEAT;
  float* f1  = w + o; o += (size_t)NN*FEAT;
  float* f2  = w + o; o += (size_t)NN*FEAT;
  float* agg = w + o; o += (size_t)NN*FEAT;
  float* gagg= w + o; o += (size_t)NN*FEAT;
  float* gA  = w + o; o += (size_t)NN*FEAT;
  float* gB  = w + o; o += (size_t)NN*FEAT;
  int*   spec = (int*)(w + o);

  auto blk = [](long n, int b) { return (unsigned)((n + b - 1) / b); };

  // ---- forward
  k_zero<<<blk(NG + 3L*NN, 256), 256, 0, stream>>>(out, NG + 3L*NN);
  k_edge_geom<<<blk(NE, 256), 256, 0, stream>>>(pos, shifts, ei, r_, u_, Y, Rb, dRb);
  k_embed<<<blk((long)NN*NC, 256), 256, 0, stream>>>(attrs, Wemb, f0, spec);

  float* feats[3] = { f0, f1, f2 };
  float* Res[2]   = { Re0, Re1 };
  for (int l = 0; l < NL; l++) {
    k_radial_fwd<<<blk(NE, 128), 256, 0, stream>>>(Rb, rW1 + l*NB*NH, rW2 + l*NH*NC, Res[l]);
    k_zero<<<blk((long)NN*FEAT, 256), 256, 0, stream>>>(agg, (long)NN*FEAT);
    k_msg_fwd<<<blk((long)NE*NC, 256), 256, 0, stream>>>(feats[l], Res[l], Y, ei, agg);
    k_node_update<<<NN, 128, 0, stream>>>(feats[l], agg, Wmix + l*NC*NC,
                                          Wskip + (long)l*NK*NC*NC, spec, feats[l+1]);
  }
  k_readout<<<blk(NN, 256), 256, 0, stream>>>(f2, attrs, ae, wro, batch, out);

  // ---- backward (forces)
  k_init_grad<<<blk((long)NN*NC, 256), 256, 0, stream>>>(wro, gA);
  k_zero<<<blk(24L*NE, 256), 256, 0, stream>>>(gY, 24L*NE);  // gY and gRb
  float* gcur = gA; float* gnext = gB;
  for (int l = NL - 1; l >= 0; l--) {
    k_node_update_bwd<<<NN, 128, 0, stream>>>(gcur, Wmix + l*NC*NC,
                                              Wskip + (long)l*NK*NC*NC, spec, gagg, gnext);
    k_msg_bwd<<<blk(NE, 4), 256, 0, stream>>>(feats[l], Res[l], Y, gagg, ei, gnext, gRe, gY);
    k_radial_bwd<<<blk(NE, 128), 256, 0, stream>>>(gRe, Rb, rW1 + l*NB*NH, rW2 + l*NH*NC, gRb);
    float* tmp = gcur; gcur = gnext; gnext = tmp;
  }
  k_force<<<blk(NE, 256), 256, 0, stream>>>(u_, r_, gY, gRb, dRb, ei, out);
}